// GCN_4097398800598
// MI455X (gfx1250) — compile-verified
//
#include <hip/hip_runtime.h>

#define HDIM 128
#define XS_STRIDE 132   // 16-row LDS tile stride (mod 64 = 4) -> conflict-free column reads

typedef float v2f __attribute__((ext_vector_type(2)));
typedef float v8f __attribute__((ext_vector_type(8)));

// ---------------- degree / symmetric norm ----------------
__global__ void k_deg_init(float* deg, int N) {
    int i = blockIdx.x * blockDim.x + threadIdx.x;
    if (i < N) deg[i] = 1.0f;                      // self-loop contribution
}
__global__ void k_deg_accum(const long long* __restrict__ dst, float* deg, int E) {
    int e = blockIdx.x * blockDim.x + threadIdx.x;
    if (e < E) atomicAdd(&deg[(int)dst[e]], 1.0f);
}
__global__ void k_deg_rsqrt(float* deg, int N) {
    int i = blockIdx.x * blockDim.x + threadIdx.x;
    if (i < N) deg[i] = rsqrtf(deg[i]);            // deg >= 1 always
}

// ---------------- layer 0: pos(Nx3) @ W0(3x128) ----------------
__global__ __launch_bounds__(256) void k_gemm_in3(const float* __restrict__ x,
                                                  const float* __restrict__ W,
                                                  float* __restrict__ h, int N) {
    int i = blockIdx.x * blockDim.x + threadIdx.x;
    if (i >= N * HDIM) return;
    int n = i >> 7, f = i & (HDIM - 1);
    const float* p = x + n * 3;
    h[i] = fmaf(p[0], W[f], fmaf(p[1], W[HDIM + f], p[2] * W[2 * HDIM + f]));
}

// ---------------- 128x128 GEMM via V_WMMA_F32_16X16X4_F32 ----------------
// Block: 256 threads = 8 waves. Block handles one 16-row tile of x (staged in LDS),
// wave w computes the 16x16 output tile at columns [16w, 16w+16).
// A (16x4 f32): lane<16 -> M=lane, K={0,1}; lane>=16 -> M=lane-16, K={2,3}
// B (4x16 f32): mirrored; C/D (16x16 f32): VGPR v, lane -> M=v+8*(lane/16), N=lane%16
__global__ __launch_bounds__(256) void k_gemm_wmma(const float* __restrict__ x,
                                                   const float* __restrict__ W,
                                                   float* __restrict__ h) {
    __shared__ float xs[16 * XS_STRIDE];
    const int tid = threadIdx.x;
    const int base = blockIdx.x * 16 * HDIM;
    // stage 16x128 input tile into LDS (2048 floats, 8 per thread)
    #pragma unroll
    for (int i = tid; i < 16 * HDIM; i += 256) {
        int r = i >> 7, c = i & (HDIM - 1);
        xs[r * XS_STRIDE + c] = x[base + i];
    }
    __syncthreads();

    const int wave  = tid >> 5;          // 0..7 -> column tile
    const int lane  = tid & 31;
    const int lhalf = lane >> 4;         // 0/1 selects K pair
    const int lmod  = lane & 15;         // M (for A) / N (for B)
    const int col   = wave * 16 + lmod;  // output column

    v8f c = {};
    #pragma unroll 4
    for (int k = 0; k < HDIM; k += 4) {
        v2f a, b;
        const int ka = k + 2 * lhalf;
        a.x = xs[lmod * XS_STRIDE + ka];
        a.y = xs[lmod * XS_STRIDE + ka + 1];
        b.x = W[ka * HDIM + col];
        b.y = W[(ka + 1) * HDIM + col];
        c = __builtin_amdgcn_wmma_f32_16x16x4_f32(false, a, false, b,
                                                  (short)0, c, false, false);
    }
    const int row0 = blockIdx.x * 16 + 8 * lhalf;
    #pragma unroll
    for (int v = 0; v < 8; ++v)
        h[(row0 + v) * HDIM + col] = c[v];
}

// ---------------- self-loop + bias init: y = b + dis^2 * h ----------------
__global__ __launch_bounds__(256) void k_init128(const float* __restrict__ h,
                                                 const float* __restrict__ dis,
                                                 const float* __restrict__ bias,
                                                 float* __restrict__ y, int N) {
    int i = blockIdx.x * blockDim.x + threadIdx.x;
    if (i >= N * HDIM) return;
    int n = i >> 7, f = i & (HDIM - 1);
    float d = dis[n];
    y[i] = fmaf(d * d, h[i], bias[f]);
}

// ---------------- edge scatter: one wave per edge, float4 per lane ----------------
__global__ __launch_bounds__(256) void k_scatter128(const long long* __restrict__ src,
                                                    const long long* __restrict__ dst,
                                                    const float* __restrict__ dis,
                                                    const float* __restrict__ h,
                                                    float* __restrict__ y, int E) {
    int warp = (blockIdx.x * 256 + threadIdx.x) >> 5;
    int lane = threadIdx.x & 31;
    if (warp >= E) return;
    int s = (int)src[warp];
    int d = (int)dst[warp];
    float w = dis[s] * dis[d];
    float4 v = ((const float4*)(h + (size_t)s * HDIM))[lane];
    float* op = y + (size_t)d * HDIM + lane * 4;
    atomicAdd(op + 0, w * v.x);
    atomicAdd(op + 1, w * v.y);
    atomicAdd(op + 2, w * v.z);
    atomicAdd(op + 3, w * v.w);
}

// ---------------- SiLU in place ----------------
__global__ __launch_bounds__(256) void k_silu(float* __restrict__ y, int total) {
    int i = blockIdx.x * blockDim.x + threadIdx.x;
    if (i >= total) return;
    float v = y[i];
    y[i] = v / (1.0f + __expf(-v));
}

// ---------------- layer 4: x(Nx128) @ W4(128x3) ----------------
__global__ __launch_bounds__(256) void k_gemm_out3(const float* __restrict__ x,
                                                   const float* __restrict__ W,
                                                   float* __restrict__ h, int N) {
    int i = blockIdx.x * blockDim.x + threadIdx.x;
    if (i >= N * 3) return;
    int n = i / 3, f = i - 3 * n;
    const float* xr = x + (size_t)n * HDIM;
    float acc = 0.f;
    #pragma unroll 8
    for (int k = 0; k < HDIM; ++k)
        acc = fmaf(xr[k], W[k * 3 + f], acc);
    h[i] = acc;
}

__global__ __launch_bounds__(256) void k_init3(const float* __restrict__ h,
                                               const float* __restrict__ dis,
                                               const float* __restrict__ bias,
                                               float* __restrict__ y, int N) {
    int i = blockIdx.x * blockDim.x + threadIdx.x;
    if (i >= N * 3) return;
    int n = i / 3, f = i - 3 * n;
    float d = dis[n];
    y[i] = fmaf(d * d, h[i], bias[f]);
}

__global__ __launch_bounds__(256) void k_scatter3(const long long* __restrict__ src,
                                                  const long long* __restrict__ dst,
                                                  const float* __restrict__ dis,
                                                  const float* __restrict__ h,
                                                  float* __restrict__ y, int E) {
    int e = blockIdx.x * blockDim.x + threadIdx.x;
    if (e >= E) return;
    int s = (int)src[e];
    int d = (int)dst[e];
    float w = dis[s] * dis[d];
    atomicAdd(&y[d * 3 + 0], w * h[s * 3 + 0]);
    atomicAdd(&y[d * 3 + 1], w * h[s * 3 + 1]);
    atomicAdd(&y[d * 3 + 2], w * h[s * 3 + 2]);
}

extern "C" void kernel_launch(void* const* d_in, const int* in_sizes, int n_in,
                              void* d_out, int out_size, void* d_ws, size_t ws_size,
                              hipStream_t stream) {
    const float*     pos  = (const float*)d_in[0];
    const long long* eidx = (const long long*)d_in[1];
    const float* W[5] = { (const float*)d_in[2], (const float*)d_in[4],
                          (const float*)d_in[6], (const float*)d_in[8],
                          (const float*)d_in[10] };
    const float* B[5] = { (const float*)d_in[3], (const float*)d_in[5],
                          (const float*)d_in[7], (const float*)d_in[9],
                          (const float*)d_in[11] };
    const int N = in_sizes[0] / 3;
    const int E = in_sizes[1] / 2;
    const long long* src = eidx;
    const long long* dst = eidx + E;

    float* ws   = (float*)d_ws;
    float* dis  = ws;                               // N
    float* bufH = ws + N;                           // N*128 (GEMM output)
    float* bufY = bufH + (size_t)N * HDIM;          // N*128 (conv output / next x)

    const int T = 256;
    const int gN    = (N + T - 1) / T;
    const int gE    = (E + T - 1) / T;
    const int gNH   = (N * HDIM + T - 1) / T;       // 25000
    const int gN3   = (N * 3 + T - 1) / T;
    const int gEdgW = (E + 7) / 8;                  // 8 edges (waves) per block
    const int gTile = N / 16;                       // 3125 (N divisible by 16)

    // symmetric normalization
    k_deg_init <<<gN, T, 0, stream>>>(dis, N);
    k_deg_accum<<<gE, T, 0, stream>>>(dst, dis, E);
    k_deg_rsqrt<<<gN, T, 0, stream>>>(dis, N);

    // layer 0: 3 -> 128
    k_gemm_in3 <<<gNH,   T, 0, stream>>>(pos, W[0], bufH, N);
    k_init128  <<<gNH,   T, 0, stream>>>(bufH, dis, B[0], bufY, N);
    k_scatter128<<<gEdgW, T, 0, stream>>>(src, dst, dis, bufH, bufY, E);
    k_silu     <<<gNH,   T, 0, stream>>>(bufY, N * HDIM);

    // layers 1..3: 128 -> 128 (WMMA)
    for (int l = 1; l <= 3; ++l) {
        k_gemm_wmma <<<gTile, T, 0, stream>>>(bufY, W[l], bufH);
        k_init128   <<<gNH,   T, 0, stream>>>(bufH, dis, B[l], bufY, N);
        k_scatter128<<<gEdgW, T, 0, stream>>>(src, dst, dis, bufH, bufY, E);
        k_silu      <<<gNH,   T, 0, stream>>>(bufY, N * HDIM);
    }

    // layer 4: 128 -> 3, into d_out
    float* out = (float*)d_out;
    k_gemm_out3<<<gN3, T, 0, stream>>>(bufY, W[4], bufH, N);
    k_init3    <<<gN3, T, 0, stream>>>(bufH, dis, B[4], out, N);
    k_scatter3 <<<gE,  T, 0, stream>>>(src, dst, dis, bufH, out, E);
}